// SAGEBC_24232205484234
// MI455X (gfx1250) — compile-verified
//
#include <hip/hip_runtime.h>
#include <hip/hip_bf16.h>

// GraphSAGE 3-layer forward for MI455X (gfx1250, wave32).
// Dense node-transform GEMMs use V_WMMA_F32_16X16X4_F32 (full f32 precision).
// Sparse mean-aggregation uses native global_atomic_add_f32 scatter.

typedef float v2f __attribute__((ext_vector_type(2)));
typedef float v8f __attribute__((ext_vector_type(8)));

#define N_NODES 50000
#define N_EDGES 600000
#define D_IN    128   // feature dim entering every aggregation (128 for all layers)

// ---------------------------------------------------------------------------
// Zero-fill (float4 granularity; count4 = number of float4 elements)
// ---------------------------------------------------------------------------
__global__ void sage_fill0(float4* __restrict__ p, int count4) {
    int i = blockIdx.x * blockDim.x + threadIdx.x;
    if (i < count4) p[i] = make_float4(0.f, 0.f, 0.f, 0.f);
}

// ---------------------------------------------------------------------------
// Degree accumulation: deg[dst] += 1 per edge
// ---------------------------------------------------------------------------
__global__ void sage_degree(const int* __restrict__ ei, float* __restrict__ deg) {
    int e = blockIdx.x * blockDim.x + threadIdx.x;
    if (e < N_EDGES) {
        int dst = ei[N_EDGES + e];
        unsafeAtomicAdd(&deg[dst], 1.0f);
    }
}

// deg -> 1 / max(deg, 1)  (in place)
__global__ void sage_invdeg(float* __restrict__ deg) {
    int i = blockIdx.x * blockDim.x + threadIdx.x;
    if (i < N_NODES) deg[i] = 1.0f / fmaxf(deg[i], 1.0f);
}

// ---------------------------------------------------------------------------
// Edge scatter: agg[dst, :] += h[src, :]   (thread = one edge x 4 features)
// ---------------------------------------------------------------------------
__global__ void sage_scatter(const float* __restrict__ h,
                             const int* __restrict__ ei,
                             float* __restrict__ agg) {
    long long tid = (long long)blockIdx.x * blockDim.x + threadIdx.x;
    if (tid >= (long long)N_EDGES * (D_IN / 4)) return;
    int e = (int)(tid >> 5);            // D_IN/4 == 32 chunks per edge
    int q = ((int)tid & 31) << 2;       // feature offset (multiple of 4)
    int src = ei[e];
    int dst = ei[N_EDGES + e];
    const float4 v = *(const float4*)(h + (long long)src * D_IN + q);
    float* a = agg + (long long)dst * D_IN + q;
    unsafeAtomicAdd(a + 0, v.x);
    unsafeAtomicAdd(a + 1, v.y);
    unsafeAtomicAdd(a + 2, v.z);
    unsafeAtomicAdd(a + 3, v.w);
}

// ---------------------------------------------------------------------------
// Fused SAGE layer GEMM:
//   out = [relu]( H @ Wself + (AGG * inv_deg) @ Wneigh + bias )
// H, AGG: [n, 128] f32.  Wself/Wneigh: [128, NCOLS].  out: [n, NCOLS].
// Block = 16-node row strip; one wave per 16x16 output tile (NCOLS/16 waves).
// K loop: 32 x V_WMMA_F32_16X16X4_F32 per matrix term.
// ---------------------------------------------------------------------------
template <int NCOLS, bool RELU>
__global__ void __launch_bounds__(32 * (NCOLS / 16))
sage_gemm_wmma(const float* __restrict__ H,
               const float* __restrict__ AGG,
               const float* __restrict__ invdeg,
               const float* __restrict__ Wself,
               const float* __restrict__ Wneigh,
               const float* __restrict__ bias,
               float* __restrict__ out) {
    // 132-float row pitch: 528 B rows => 16B-aligned float4 stores,
    // and column reads hit distinct LDS banks (4*m mod 64 unique for m<16).
    __shared__ float sSelf[16][132];
    __shared__ float sMean[16][132];

    const int t        = threadIdx.x;
    const int nthreads = 32 * (NCOLS / 16);
    const int rowbase  = blockIdx.x * 16;

    // Cooperative stage of the 16x128 H tile and mean tile into LDS.
    for (int i = t; i < 16 * (D_IN / 4); i += nthreads) {
        int r  = i >> 5;          // row 0..15
        int c4 = (i & 31) << 2;   // col 0,4,...,124
        long long gbase = (long long)(rowbase + r) * D_IN + c4;
        float4 hv = *(const float4*)(H + gbase);
        float4 av = *(const float4*)(AGG + gbase);
        float id  = invdeg[rowbase + r];
        *(float4*)&sSelf[r][c4] = hv;
        *(float4*)&sMean[r][c4] = make_float4(av.x * id, av.y * id, av.z * id, av.w * id);
    }
    __syncthreads();

    const int lane = t & 31;
    const int wave = t >> 5;
    const int m    = lane & 15;            // A-fragment row
    const int koff = (lane >> 4) << 1;     // A/B-fragment K sub-offset {0,2}
    const int n    = wave * 16 + (lane & 15);  // output column

    v8f acc = {};

    #pragma unroll 4
    for (int k0 = 0; k0 < D_IN; k0 += 4) {
        const int ka = k0 + koff;
        v2f aS, aM, bS, bM;
        aS.x = sSelf[m][ka];
        aS.y = sSelf[m][ka + 1];
        aM.x = sMean[m][ka];
        aM.y = sMean[m][ka + 1];
        bS.x = Wself[(long long)ka * NCOLS + n];
        bS.y = Wself[(long long)(ka + 1) * NCOLS + n];
        bM.x = Wneigh[(long long)ka * NCOLS + n];
        bM.y = Wneigh[(long long)(ka + 1) * NCOLS + n];
        acc = __builtin_amdgcn_wmma_f32_16x16x4_f32(
            false, aS, false, bS, (short)0, acc, false, false);
        acc = __builtin_amdgcn_wmma_f32_16x16x4_f32(
            false, aM, false, bM, (short)0, acc, false, false);
    }

    const float bv   = bias[n];
    const int   rsub = (lane >> 4) << 3;   // 0 or 8
    #pragma unroll
    for (int v = 0; v < 8; ++v) {
        float val = acc[v] + bv;
        if (RELU) val = fmaxf(val, 0.f);
        out[(long long)(rowbase + v + rsub) * NCOLS + n] = val;
    }
}

// ---------------------------------------------------------------------------
// Host-side orchestration (all on `stream`, graph-capture safe)
// ---------------------------------------------------------------------------
extern "C" void kernel_launch(void* const* d_in, const int* in_sizes, int n_in,
                              void* d_out, int out_size, void* d_ws, size_t ws_size,
                              hipStream_t stream) {
    (void)in_sizes; (void)n_in; (void)out_size; (void)ws_size;

    const float* x   = (const float*)d_in[0];
    const int*   ei  = (const int*)  d_in[1];
    const float* Ws0 = (const float*)d_in[2];
    const float* Wn0 = (const float*)d_in[3];
    const float* b0  = (const float*)d_in[4];
    const float* Ws1 = (const float*)d_in[5];
    const float* Wn1 = (const float*)d_in[6];
    const float* b1  = (const float*)d_in[7];
    const float* Ws2 = (const float*)d_in[8];
    const float* Wn2 = (const float*)d_in[9];
    const float* b2  = (const float*)d_in[10];
    float* out = (float*)d_out;

    // Workspace layout (floats): inv_deg | agg | h1 | h2
    float* inv_deg = (float*)d_ws;
    float* agg = inv_deg + 50176;                       // 50000 padded to 256
    float* h1  = agg + (size_t)N_NODES * D_IN;
    float* h2  = h1  + (size_t)N_NODES * D_IN;

    const int TB = 256;
    const int aggN4   = N_NODES * D_IN / 4;
    const int scatterN = N_EDGES * (D_IN / 4);

    // --- degrees ---
    sage_fill0<<<(N_NODES / 4 + TB - 1) / TB, TB, 0, stream>>>((float4*)inv_deg, N_NODES / 4);
    sage_degree<<<(N_EDGES + TB - 1) / TB, TB, 0, stream>>>(ei, inv_deg);
    sage_invdeg<<<(N_NODES + TB - 1) / TB, TB, 0, stream>>>(inv_deg);

    // --- layer 0: x -> h1 (relu) ---
    sage_fill0<<<(aggN4 + TB - 1) / TB, TB, 0, stream>>>((float4*)agg, aggN4);
    sage_scatter<<<(scatterN + TB - 1) / TB, TB, 0, stream>>>(x, ei, agg);
    sage_gemm_wmma<128, true><<<N_NODES / 16, 256, 0, stream>>>(
        x, agg, inv_deg, Ws0, Wn0, b0, h1);

    // --- layer 1: h1 -> h2 (relu) ---
    sage_fill0<<<(aggN4 + TB - 1) / TB, TB, 0, stream>>>((float4*)agg, aggN4);
    sage_scatter<<<(scatterN + TB - 1) / TB, TB, 0, stream>>>(h1, ei, agg);
    sage_gemm_wmma<128, true><<<N_NODES / 16, 256, 0, stream>>>(
        h1, agg, inv_deg, Ws1, Wn1, b1, h2);

    // --- layer 2: h2 -> out (no relu, NCOLS=64) ---
    sage_fill0<<<(aggN4 + TB - 1) / TB, TB, 0, stream>>>((float4*)agg, aggN4);
    sage_scatter<<<(scatterN + TB - 1) / TB, TB, 0, stream>>>(h2, ei, agg);
    sage_gemm_wmma<64, false><<<N_NODES / 16, 128, 0, stream>>>(
        h2, agg, inv_deg, Ws2, Wn2, b2, out);
}